// PermutationEquivariantLayer_11948599018284
// MI455X (gfx1250) — compile-verified
//
#include <hip/hip_runtime.h>

typedef __attribute__((ext_vector_type(2))) float v2f;
typedef __attribute__((ext_vector_type(8))) float v8f;

#define B_  64
#define N_  32768
#define C_  2
#define D1_ 32
#define D2_ 64

// ---------------------------------------------------------------------------
// Kernel 1: per-batch max over the N axis.  beta[b][c] = max_n x[b][n][c]
// ---------------------------------------------------------------------------
__global__ __launch_bounds__(256) void k_batch_max(const float* __restrict__ x,
                                                   float* __restrict__ beta) {
  __shared__ float s0[256];
  __shared__ float s1[256];
  const int b   = blockIdx.x;
  const int tid = threadIdx.x;
  const float* xb = x + (size_t)b * N_ * C_;
  float m0 = -3.402823466e38f, m1 = -3.402823466e38f;
  for (int n = tid; n < N_; n += 256) {
    m0 = fmaxf(m0, xb[2 * n]);
    m1 = fmaxf(m1, xb[2 * n + 1]);
  }
  s0[tid] = m0; s1[tid] = m1;
  __syncthreads();
  for (int off = 128; off > 0; off >>= 1) {
    if (tid < off) {
      s0[tid] = fmaxf(s0[tid], s0[tid + off]);
      s1[tid] = fmaxf(s1[tid], s1[tid + off]);
    }
    __syncthreads();
  }
  if (tid == 0) { beta[2 * b] = s0[0]; beta[2 * b + 1] = s1[0]; }
}

// ---------------------------------------------------------------------------
// Kernel 2: fold the two linear layers.
//   blocks 0..63 : cvec[b][j] = b2[j] + sum_k (b1[k]-beta0*G1[0][k]-beta1*G1[1][k]) * W2[k][j]
//   block  64    : M[c][j]    = sum_k W1[c][k] * W2[k][j]     (c = 0,1)
// ---------------------------------------------------------------------------
__global__ __launch_bounds__(64) void k_prep(const float* __restrict__ W1,
                                             const float* __restrict__ b1,
                                             const float* __restrict__ G1,
                                             const float* __restrict__ W2,
                                             const float* __restrict__ b2,
                                             const float* __restrict__ beta,
                                             float* __restrict__ cvec,
                                             float* __restrict__ Mmat) {
  const int j  = threadIdx.x;      // 0..63  (output column)
  const int bb = blockIdx.x;
  if (bb < B_) {
    const float be0 = beta[2 * bb], be1 = beta[2 * bb + 1];
    float acc = b2[j];
    for (int k = 0; k < D1_; ++k) {
      float t = b1[k] - be0 * G1[k] - be1 * G1[D1_ + k];
      acc += t * W2[k * D2_ + j];
    }
    cvec[bb * D2_ + j] = acc;
  } else {
    float m0 = 0.f, m1 = 0.f;
    for (int k = 0; k < D1_; ++k) {
      const float w2 = W2[k * D2_ + j];
      m0 += W1[k]        * w2;
      m1 += W1[D1_ + k]  * w2;
    }
    Mmat[j]        = m0;
    Mmat[D2_ + j]  = m1;
  }
}

// ---------------------------------------------------------------------------
// Kernel 3: streaming fused layer via V_WMMA_F32_16X16X4_F32.
//   out[b, p, j] = x[b,p,0]*M[0][j] + x[b,p,1]*M[1][j] + cvec[b][j]
// A (16x4 f32): lanes 0-15 carry K0/K1 of 16 points, lanes 16-31 carry the
// zero K2/K3 pad.  B (4x16): M columns in lanes 0-15, zero pad in 16-31.
// C: cvec broadcast down each column.  Non-temporal stores (536 MB stream).
// ---------------------------------------------------------------------------
__global__ __launch_bounds__(256) void k_main(const float* __restrict__ x,
                                              const float* __restrict__ Mmat,
                                              const float* __restrict__ cvec,
                                              float* __restrict__ out) {
  const int b    = blockIdx.y;
  const int lane = threadIdx.x & 31;
  const int wave = threadIdx.x >> 5;
  const bool lo  = lane < 16;
  const int col  = lane & 15;

  // Per-wave constant B tiles and bias values (one per 16-column tile).
  v2f  btile[4];
  float cv[4];
#pragma unroll
  for (int t = 0; t < 4; ++t) {
    btile[t].x = lo ? Mmat[t * 16 + col]        : 0.0f;   // K=0 row (K=2 pad = 0)
    btile[t].y = lo ? Mmat[D2_ + t * 16 + col]  : 0.0f;   // K=1 row (K=3 pad = 0)
    cv[t]      = cvec[b * D2_ + t * 16 + col];            // same for both lane halves
  }

  const float* xb = x + (size_t)b * N_ * C_;
  const int chunk0 = blockIdx.x * 64 + wave;   // 64 chunks of 16 points per block

  for (int i = 0; i < 8; ++i) {
    const int p0 = (chunk0 + i * 8) * 16;

    v2f a;
    if (lo) {
      a = *(const v2f*)(xb + (size_t)(p0 + col) * 2);     // point features (K0,K1)
    } else {
      a.x = 0.0f; a.y = 0.0f;                             // K2,K3 zero pad
    }

#pragma unroll
    for (int t = 0; t < 4; ++t) {
      v8f c;
#pragma unroll
      for (int r = 0; r < 8; ++r) c[r] = cv[t];

      v8f d = __builtin_amdgcn_wmma_f32_16x16x4_f32(
          /*neg_a=*/false, a, /*neg_b=*/false, btile[t],
          /*c_mod=*/(short)0, c, /*reuse_a=*/false, /*reuse_b=*/false);

      // D layout: VGPR r -> row (r or r+8), column = lane&15 within tile t.
      const int rowbase = p0 + (lo ? 0 : 8);
      float* obase = out + ((size_t)b * N_ + rowbase) * D2_ + t * 16 + col;
#pragma unroll
      for (int r = 0; r < 8; ++r)
        __builtin_nontemporal_store(d[r], obase + (size_t)r * D2_);
    }
  }
}

// ---------------------------------------------------------------------------
extern "C" void kernel_launch(void* const* d_in, const int* in_sizes, int n_in,
                              void* d_out, int out_size, void* d_ws, size_t ws_size,
                              hipStream_t stream) {
  const float* x  = (const float*)d_in[0];   // [64, 32768, 2]
  const float* W1 = (const float*)d_in[1];   // [2, 32]
  const float* b1 = (const float*)d_in[2];   // [32]
  const float* G1 = (const float*)d_in[3];   // [2, 32]
  const float* W2 = (const float*)d_in[4];   // [32, 64]
  const float* b2 = (const float*)d_in[5];   // [64]
  float* out = (float*)d_out;                // [64, 32768, 64]

  float* ws   = (float*)d_ws;
  float* beta = ws;                          // [64][2]
  float* cvec = ws + 128;                    // [64][64]
  float* Mmat = ws + 128 + B_ * D2_;         // [2][64]

  k_batch_max<<<B_, 256, 0, stream>>>(x, beta);
  k_prep<<<B_ + 1, 64, 0, stream>>>(W1, b1, G1, W2, b2, beta, cvec, Mmat);
  k_main<<<dim3(N_ / (64 * 16), B_), 256, 0, stream>>>(x, Mmat, cvec, out);
}